// TRED_GNN4_59210419142916
// MI455X (gfx1250) — compile-verified
//
#include <hip/hip_runtime.h>
#include <math.h>

#define L_LAYERS 3
#define E_EDGES  1000000
#define N_NODES  100000
#define D_DIM    64
#define A_DIM    8
#define NR2_DIM  461
#define Q_DIM    1000
#define ENT_DIM  10000

typedef __attribute__((ext_vector_type(2))) float v2f;
typedef __attribute__((ext_vector_type(8))) float v8f;

// D = A(16x4, f32) * B(4x16, f32) + C(16x16, f32) -- CDNA5 wave32 WMMA
__device__ __forceinline__ v8f wmma4(v2f a, v2f b, v8f c) {
  return __builtin_amdgcn_wmma_f32_16x16x4_f32(false, a, false, b, (short)0, c,
                                               false, false);
}

__device__ __forceinline__ float sigmoidf_(float x) {
  return 1.0f / (1.0f + __expf(-x));
}

__global__ void zero_kernel(float* __restrict__ p, int n) {
  int i = blockIdx.x * blockDim.x + threadIdx.x;
  int stride = gridDim.x * blockDim.x;
  for (; i < n; i += stride) p[i] = 0.0f;
}

// ---------------------------------------------------------------------------
// Edge phase: 8 lanes per edge, each lane owns 8 of the 64 feature channels.
// attn = relu(hs@Ws^T + hr@Wr^T + hqr@Wqr^T + b); alpha = sigmoid(attn@wal+b)
// agg[obj] += alpha * (hs + hr)   (f32 global atomics)
// ---------------------------------------------------------------------------
__global__ void __launch_bounds__(256)
edge_kernel(const int* __restrict__ sub, const int* __restrict__ rel,
            const int* __restrict__ r_idx, const int* __restrict__ relation,
            const int* __restrict__ obj,
            const float* __restrict__ hstate, const float* __restrict__ emb,
            const float* __restrict__ Ws, const float* __restrict__ Wr,
            const float* __restrict__ Wqr, const float* __restrict__ Wqr_b,
            const float* __restrict__ wal_w, const float* __restrict__ wal_b,
            float* __restrict__ agg) {
  __shared__ float sW[3 * A_DIM * D_DIM];  // Ws | Wr | Wqr
  __shared__ float sB[A_DIM], sAl[A_DIM], sAlB[1];
  for (int t = threadIdx.x; t < A_DIM * D_DIM; t += blockDim.x) {
    sW[t]                    = Ws[t];
    sW[t + A_DIM * D_DIM]    = Wr[t];
    sW[t + 2 * A_DIM * D_DIM]= Wqr[t];
  }
  if (threadIdx.x < A_DIM) {
    sB[threadIdx.x]  = Wqr_b[threadIdx.x];
    sAl[threadIdx.x] = wal_w[threadIdx.x];
  }
  if (threadIdx.x == 0) sAlB[0] = wal_b[0];
  __syncthreads();

  int gid = blockIdx.x * blockDim.x + threadIdx.x;
  int e = gid >> 3;
  if (e >= E_EDGES) return;
  int sl = gid & 7;
  int ks = sl * 8;

  int vs = sub[e];
  int vr = rel[e];
  int vq = relation[r_idx[e]];
  int vo = obj[e];

  const float* ps = hstate + (size_t)vs * D_DIM + ks;
  const float* pr = emb + (size_t)vr * D_DIM + ks;
  const float* pq = emb + (size_t)vq * D_DIM + ks;
  float hs[8], hr[8], hq[8];
#pragma unroll
  for (int t = 0; t < 8; ++t) { hs[t] = ps[t]; hr[t] = pr[t]; hq[t] = pq[t]; }

  float p[A_DIM];
#pragma unroll
  for (int a = 0; a < A_DIM; ++a) {
    const float* w0 = sW + a * D_DIM + ks;
    const float* w1 = w0 + A_DIM * D_DIM;
    const float* w2 = w0 + 2 * A_DIM * D_DIM;
    float acc = 0.0f;
#pragma unroll
    for (int t = 0; t < 8; ++t)
      acc = fmaf(hs[t], w0[t], fmaf(hr[t], w1[t], fmaf(hq[t], w2[t], acc)));
    p[a] = acc;
  }
  // butterfly reduce across the 8 lanes of this edge group
#pragma unroll
  for (int m = 1; m < 8; m <<= 1) {
#pragma unroll
    for (int a = 0; a < A_DIM; ++a) p[a] += __shfl_xor(p[a], m, 8);
  }

  float dal = 0.0f;
#pragma unroll
  for (int a = 0; a < A_DIM; ++a)
    dal = fmaf(fmaxf(p[a] + sB[a], 0.0f), sAl[a], dal);
  float alpha = sigmoidf_(dal + sAlB[0]);

  float* dst = agg + (size_t)vo * D_DIM + ks;
#pragma unroll
  for (int t = 0; t < 8; ++t) atomicAdd(dst + t, alpha * (hs[t] + hr[t]));
}

// ---------------------------------------------------------------------------
// hidden = relu(agg @ Wh^T)  -- [N,64]x[64,64] via WMMA f32 16x16x4
// One wave per 16-row tile; 4 column tiles; K=64 in 16 steps of 4.
// ---------------------------------------------------------------------------
__global__ void __launch_bounds__(256)
wh_relu_kernel(const float* __restrict__ agg, const float* __restrict__ Wh,
               float* __restrict__ hidden) {
  int lane = threadIdx.x & 31;
  int wv = threadIdx.x >> 5;
  int tile = blockIdx.x * 8 + wv;
  if (tile * 16 >= N_NODES) return;  // wave-uniform exit: EXEC stays all-1s
  int row0 = tile * 16;
  int mrow = lane & 15;
  int khalf = (lane >> 4) << 1;  // lanes 0-15 hold K0/K1, lanes 16-31 K2/K3

  v2f a[16];
  const float* arow = agg + (size_t)(row0 + mrow) * D_DIM + khalf;
#pragma unroll
  for (int s = 0; s < 16; ++s) a[s] = *(const v2f*)(arow + s * 4);

#pragma unroll
  for (int ct = 0; ct < 4; ++ct) {
    v8f acc = {};
    const float* brow = Wh + (size_t)(ct * 16 + (lane & 15)) * D_DIM + khalf;
#pragma unroll
    for (int s = 0; s < 16; ++s) {
      v2f b = *(const v2f*)(brow + s * 4);
      acc = wmma4(a[s], b, acc);
    }
    int col = ct * 16 + (lane & 15);
    int mbase = row0 + ((lane >> 4) << 3);
#pragma unroll
    for (int v = 0; v < 8; ++v)
      hidden[(size_t)(mbase + v) * D_DIM + col] = fmaxf(acc[v], 0.0f);
  }
}

// ---------------------------------------------------------------------------
// Fused GRU cell: gi = x@Wih^T+bih, gh = h@Whh^T+bhh (both [N,192] via WMMA),
// gates computed in the C-matrix layout, in-place h update.
// ---------------------------------------------------------------------------
__global__ void __launch_bounds__(256)
gru_kernel(const float* __restrict__ x, const float* __restrict__ Wih,
           const float* __restrict__ Whh, const float* __restrict__ bih,
           const float* __restrict__ bhh, float* __restrict__ h) {
  int lane = threadIdx.x & 31;
  int wv = threadIdx.x >> 5;
  int tile = blockIdx.x * 8 + wv;
  if (tile * 16 >= N_NODES) return;
  int row0 = tile * 16;
  int mrow = lane & 15;
  int khalf = (lane >> 4) << 1;

  v2f ax[16], ah[16];
  const float* xrow = x + (size_t)(row0 + mrow) * D_DIM + khalf;
  const float* hrow = h + (size_t)(row0 + mrow) * D_DIM + khalf;
#pragma unroll
  for (int s = 0; s < 16; ++s) {
    ax[s] = *(const v2f*)(xrow + s * 4);
    ah[s] = *(const v2f*)(hrow + s * 4);
  }

#pragma unroll
  for (int ct = 0; ct < 4; ++ct) {
    int j = ct * 16 + (lane & 15);  // output column in [0,64)
    const float* wir = Wih + (size_t)j * D_DIM + khalf;
    const float* wiz = wir + 64 * D_DIM;
    const float* win = wir + 128 * D_DIM;
    const float* whr = Whh + (size_t)j * D_DIM + khalf;
    const float* whz = whr + 64 * D_DIM;
    const float* whn = whr + 128 * D_DIM;
    v8f air = {}, ahr = {}, aiz = {}, ahz = {}, ain = {}, ahn = {};
#pragma unroll
    for (int s = 0; s < 16; ++s) {
      int ko = s * 4;
      air = wmma4(ax[s], *(const v2f*)(wir + ko), air);
      ahr = wmma4(ah[s], *(const v2f*)(whr + ko), ahr);
      aiz = wmma4(ax[s], *(const v2f*)(wiz + ko), aiz);
      ahz = wmma4(ah[s], *(const v2f*)(whz + ko), ahz);
      ain = wmma4(ax[s], *(const v2f*)(win + ko), ain);
      ahn = wmma4(ah[s], *(const v2f*)(whn + ko), ahn);
    }
    float b_r  = bih[j] + bhh[j];
    float b_z  = bih[64 + j] + bhh[64 + j];
    float b_in = bih[128 + j];
    float b_hn = bhh[128 + j];
    int mbase = row0 + ((lane >> 4) << 3);
#pragma unroll
    for (int v = 0; v < 8; ++v) {
      size_t off = (size_t)(mbase + v) * D_DIM + j;
      float h0v = h[off];
      float rg = sigmoidf_(air[v] + ahr[v] + b_r);
      float zg = sigmoidf_(aiz[v] + ahz[v] + b_z);
      float ng = tanhf(ain[v] + b_in + rg * (ahn[v] + b_hn));
      h[off] = (1.0f - zg) * ng + zg * h0v;
    }
  }
}

// ---------------------------------------------------------------------------
// scores = h @ Wfinal^T ; scatter into [Q, ENT]
// ---------------------------------------------------------------------------
__global__ void __launch_bounds__(256)
final_kernel(const float* __restrict__ h, const float* __restrict__ Wfin,
             const int* __restrict__ nodes_q, const int* __restrict__ nodes_ent,
             float* __restrict__ out) {
  int gid = blockIdx.x * blockDim.x + threadIdx.x;
  int node = gid >> 3;
  if (node >= N_NODES) return;
  int sl = gid & 7;
  const float* ph = h + (size_t)node * D_DIM + sl * 8;
  const float* pw = Wfin + sl * 8;
  float acc = 0.0f;
#pragma unroll
  for (int t = 0; t < 8; ++t) acc = fmaf(ph[t], pw[t], acc);
#pragma unroll
  for (int m = 1; m < 8; m <<= 1) acc += __shfl_xor(acc, m, 8);
  if (sl == 0)
    out[(size_t)nodes_q[node] * ENT_DIM + nodes_ent[node]] = acc;
}

extern "C" void kernel_launch(void* const* d_in, const int* in_sizes, int n_in,
                              void* d_out, int out_size, void* d_ws, size_t ws_size,
                              hipStream_t stream) {
  const int*   relation = (const int*)d_in[0];
  const int*   r_idx    = (const int*)d_in[1];
  const int*   rel      = (const int*)d_in[2];
  const int*   sub      = (const int*)d_in[3];
  const int*   obj      = (const int*)d_in[4];
  // d_in[5] = idx (arange -> identity index_copy, unused)
  const int*   nodes_q  = (const int*)d_in[6];
  const int*   nodes_ent= (const int*)d_in[7];
  const float* rela     = (const float*)d_in[8];
  const float* Ws       = (const float*)d_in[9];
  const float* Wr       = (const float*)d_in[10];
  const float* Wqr      = (const float*)d_in[11];
  const float* Wqrb     = (const float*)d_in[12];
  const float* walw     = (const float*)d_in[13];
  const float* walb     = (const float*)d_in[14];
  const float* Wh       = (const float*)d_in[15];
  const float* Wih      = (const float*)d_in[16];
  const float* Whh      = (const float*)d_in[17];
  const float* bih      = (const float*)d_in[18];
  const float* bhh      = (const float*)d_in[19];
  const float* Wfin     = (const float*)d_in[20];
  float* out = (float*)d_out;

  float* h0     = (float*)d_ws;                       // [N,64] state
  float* hidden = h0 + (size_t)N_NODES * D_DIM;       // [N,64] intermediate
  float* agg    = hidden + (size_t)N_NODES * D_DIM;   // [N,64] scatter target

  const int nd = N_NODES * D_DIM;
  zero_kernel<<<2048, 256, 0, stream>>>(h0, nd);

  const int egrid = (E_EDGES * 8 + 255) / 256;
  const int tiles = N_NODES / 16;            // 6250
  const int nblk  = (tiles + 7) / 8;         // 8 waves/block

  for (int i = 0; i < L_LAYERS; ++i) {
    zero_kernel<<<2048, 256, 0, stream>>>(agg, nd);
    edge_kernel<<<egrid, 256, 0, stream>>>(
        sub + (size_t)i * E_EDGES, rel + (size_t)i * E_EDGES,
        r_idx + (size_t)i * E_EDGES, relation, obj + (size_t)i * E_EDGES,
        h0, rela + (size_t)i * NR2_DIM * D_DIM,
        Ws + (size_t)i * A_DIM * D_DIM, Wr + (size_t)i * A_DIM * D_DIM,
        Wqr + (size_t)i * A_DIM * D_DIM, Wqrb + (size_t)i * A_DIM,
        walw + (size_t)i * A_DIM, walb + i, agg);
    wh_relu_kernel<<<nblk, 256, 0, stream>>>(agg, Wh + (size_t)i * D_DIM * D_DIM,
                                             hidden);
    gru_kernel<<<nblk, 256, 0, stream>>>(hidden, Wih, Whh, bih, bhh, h0);
  }

  zero_kernel<<<4096, 256, 0, stream>>>(out, Q_DIM * ENT_DIM);
  final_kernel<<<(N_NODES * 8 + 255) / 256, 256, 0, stream>>>(
      h0, Wfin, nodes_q, nodes_ent, out);
}